// TemporalBiasedAttention_32933809226033
// MI455X (gfx1250) — compile-verified
//
#include <hip/hip_runtime.h>
#include <hip/hip_bf16.h>

#define B_ 32
#define T_ 2048
#define C_ 1024
#define H_ 8
#define D_ 128

typedef __attribute__((ext_vector_type(2))) float v2f;
typedef __attribute__((ext_vector_type(8))) float v8f;

// ---------------- Phase 0a: q[i] = q_w[i,:] . query_token + q_b[i] ----------------
__global__ void q_kernel(const float* __restrict__ qt, const float* __restrict__ q_w,
                         const float* __restrict__ q_b, float* __restrict__ q) {
    const int o    = (blockIdx.x * 256 + threadIdx.x) >> 5;   // 0..1023
    const int lane = threadIdx.x & 31;
    const float* wr = q_w + (size_t)o * C_;
    float acc = 0.f;
    for (int cb = lane * 4; cb < C_; cb += 128) {
        const float4 a = *(const float4*)(qt + cb);
        const float4 w = *(const float4*)(wr + cb);
        acc = fmaf(a.x, w.x, acc); acc = fmaf(a.y, w.y, acc);
        acc = fmaf(a.z, w.z, acc); acc = fmaf(a.w, w.w, acc);
    }
    #pragma unroll
    for (int s = 16; s > 0; s >>= 1) acc += __shfl_xor(acc, s, 32);
    if (lane == 0) q[o] = acc + q_b[o];
}

// ------- Phase 0b: qk_vec[h,c] = sum_d q[h,d]*kv_w_k[h*D+d, c]; const_k[h] -------
__global__ void qkvec_kernel(const float* __restrict__ q, const float* __restrict__ kv_w,
                             const float* __restrict__ kv_b, float* __restrict__ qk_vec,
                             float* __restrict__ const_k) {
    const int idx = blockIdx.x * 256 + threadIdx.x;   // h*C + c
    const int h = idx / C_, c = idx % C_;
    float acc = 0.f;
    for (int d = 0; d < D_; ++d)
        acc = fmaf(q[h * D_ + d], kv_w[(size_t)(h * D_ + d) * C_ + c], acc);
    qk_vec[idx] = acc;
    if (c == 0) {
        float ck = 0.f;
        for (int d = 0; d < D_; ++d) ck = fmaf(q[h * D_ + d], kv_b[h * D_ + d], ck);
        const_k[h] = ck;
    }
}

// ------- Phase 1: logits[b,h,t] = scale*(x[b,t,:].qk_vec[h] + const_k[h]) + bias(t) -------
__global__ void logits_kernel(const float* __restrict__ x, const int* __restrict__ mask,
                              const float* __restrict__ qk_vec, const float* __restrict__ const_k,
                              float* __restrict__ logits) {
    __shared__ float qk[H_][C_];   // 32 KB
    const int b  = blockIdx.y;
    const int t0 = blockIdx.x * 64;
    for (int i = threadIdx.x; i < H_ * C_; i += 256) qk[i / C_][i % C_] = qk_vec[i];
    __syncthreads();
    const int wave = threadIdx.x >> 5;
    const int lane = threadIdx.x & 31;
    const float scale = 0.088388347648318447f;  // 1/sqrt(128)
    for (int r = 0; r < 8; ++r) {
        const int t = t0 + wave * 8 + r;
        const float* xrow = x + ((size_t)b * T_ + t) * C_;
        float acc[H_];
        #pragma unroll
        for (int h = 0; h < H_; ++h) acc[h] = 0.f;
        for (int cb = 0; cb < C_; cb += 128) {
            const float4 xv = *(const float4*)(xrow + cb + lane * 4);
            #pragma unroll
            for (int h = 0; h < H_; ++h) {
                const float4 qv = *(const float4*)(&qk[h][cb + lane * 4]);
                acc[h] = fmaf(xv.x, qv.x, acc[h]);
                acc[h] = fmaf(xv.y, qv.y, acc[h]);
                acc[h] = fmaf(xv.z, qv.z, acc[h]);
                acc[h] = fmaf(xv.w, qv.w, acc[h]);
            }
        }
        #pragma unroll
        for (int h = 0; h < H_; ++h) {
            #pragma unroll
            for (int s = 16; s > 0; s >>= 1) acc[h] += __shfl_xor(acc[h], s, 32);
        }
        if (lane < H_) {
            const float bias = 2.0f * __expf(-(float)t / ((float)T_ / 3.0f));
            float v = (acc[lane] + const_k[lane]) * scale + bias;
            if (mask[(size_t)b * T_ + t] == 0) v = -__builtin_inff();
            logits[((size_t)b * H_ + lane) * T_ + t] = v;
        }
    }
}

// ------- Phase 2: softmax over T per (b,h), in place -------
__global__ void softmax_kernel(float* __restrict__ attn) {
    const int bh = blockIdx.x;
    float* row = attn + (size_t)bh * T_;
    __shared__ float red[256];
    float v[8];
    float m = -__builtin_inff();
    #pragma unroll
    for (int i = 0; i < 8; ++i) { v[i] = row[threadIdx.x + i * 256]; m = fmaxf(m, v[i]); }
    red[threadIdx.x] = m; __syncthreads();
    for (int s = 128; s > 0; s >>= 1) {
        if (threadIdx.x < s) red[threadIdx.x] = fmaxf(red[threadIdx.x], red[threadIdx.x + s]);
        __syncthreads();
    }
    m = red[0]; __syncthreads();
    float sum = 0.f;
    #pragma unroll
    for (int i = 0; i < 8; ++i) { v[i] = __expf(v[i] - m); sum += v[i]; }
    red[threadIdx.x] = sum; __syncthreads();
    for (int s = 128; s > 0; s >>= 1) {
        if (threadIdx.x < s) red[threadIdx.x] += red[threadIdx.x + s];
        __syncthreads();
    }
    const float inv = 1.0f / red[0];
    #pragma unroll
    for (int i = 0; i < 8; ++i) row[threadIdx.x + i * 256] = v[i] * inv;
}

// ------- Phase 3 (WMMA): xp[b,h,:] = attn[b,h,:] @ x[b,:,:]  (M=16 pad, K=T, N=C) -------
#define KC 64                   // K chunk
#define NT 128                  // N tile per block (8 waves x 16)
#define XS (NT + 4)             // padded LDS stride
__global__ void xp_wmma_kernel(const float* __restrict__ attn, const float* __restrict__ x,
                               float* __restrict__ xp) {
    __shared__ float aTile[16][KC + 4];   // A: heads x K chunk (rows 8..15 = 0)
    __shared__ float xTile[KC][XS];       // B: K chunk x N tile
    const int b    = blockIdx.x;
    const int n0   = blockIdx.y * NT;
    const int wave = threadIdx.x >> 5;
    const int lane = threadIdx.x & 31;
    const int m    = lane & 15;               // A row (head) / B col within sub-tile
    const int koff = (lane >> 4) << 1;        // 0 for lanes 0-15, 2 for lanes 16-31
    v8f acc = {};
    for (int k0 = 0; k0 < T_; k0 += KC) {
        for (int i = threadIdx.x; i < 16 * KC; i += 256) {
            const int hh = i / KC, kk = i % KC;
            aTile[hh][kk] = (hh < H_) ? attn[((size_t)b * H_ + hh) * T_ + k0 + kk] : 0.0f;
        }
        for (int rr = wave; rr < KC; rr += 8) {
            const float4 xv = *(const float4*)(x + ((size_t)b * T_ + k0 + rr) * C_ + n0 + lane * 4);
            *(float4*)(&xTile[rr][lane * 4]) = xv;
        }
        __syncthreads();
        #pragma unroll 4
        for (int kk = 0; kk < KC; kk += 4) {
            v2f af, bf;
            af.x = aTile[m][kk + koff];
            af.y = aTile[m][kk + koff + 1];
            bf.x = xTile[kk + koff][wave * 16 + m];
            bf.y = xTile[kk + koff + 1][wave * 16 + m];
            acc = __builtin_amdgcn_wmma_f32_16x16x4_f32(false, af, false, bf,
                                                        (short)0, acc, false, false);
        }
        __syncthreads();
    }
    if (lane < 16) {
        const int ncol = n0 + wave * 16 + lane;
        #pragma unroll
        for (int r = 0; r < H_; ++r)   // D rows 0..7 = heads; rows 8..15 are pad
            xp[((size_t)b * H_ + r) * C_ + ncol] = acc[r];
    }
}

// ------- Phase 4: pooled[b, h*D+d] = xp[b,h,:] . kv_w_v[h*D+d,:] + kv_b_v -------
__global__ void pooled_kernel(const float* __restrict__ xp, const float* __restrict__ kv_w,
                              const float* __restrict__ kv_b, float* __restrict__ pooled) {
    const int o    = (blockIdx.x * 256 + threadIdx.x) >> 5;   // 0..B*C-1
    const int lane = threadIdx.x & 31;
    const int bo = o / C_;
    const int i  = o % C_;          // h*D + d
    const int h  = i / D_;
    const float* xr = xp + ((size_t)bo * H_ + h) * C_;
    const float* wr = kv_w + (size_t)(C_ + i) * C_;   // v half of kv_w
    float acc = 0.f;
    for (int cb = lane * 4; cb < C_; cb += 128) {
        const float4 a = *(const float4*)(xr + cb);
        const float4 w = *(const float4*)(wr + cb);
        acc = fmaf(a.x, w.x, acc); acc = fmaf(a.y, w.y, acc);
        acc = fmaf(a.z, w.z, acc); acc = fmaf(a.w, w.w, acc);
    }
    #pragma unroll
    for (int s = 16; s > 0; s >>= 1) acc += __shfl_xor(acc, s, 32);
    if (lane == 0) pooled[(size_t)bo * C_ + i] = acc + kv_b[C_ + i];
}

// ------- Phase 5: out[b,j] = pooled[b,:] . proj_w[j,:] + proj_b[j] -------
__global__ void proj_kernel(const float* __restrict__ pooled, const float* __restrict__ proj_w,
                            const float* __restrict__ proj_b, float* __restrict__ out) {
    const int o    = (blockIdx.x * 256 + threadIdx.x) >> 5;
    const int lane = threadIdx.x & 31;
    const int bo = o / C_;
    const int j  = o % C_;
    const float* pr = pooled + (size_t)bo * C_;
    const float* wr = proj_w + (size_t)j * C_;
    float acc = 0.f;
    for (int cb = lane * 4; cb < C_; cb += 128) {
        const float4 a = *(const float4*)(pr + cb);
        const float4 w = *(const float4*)(wr + cb);
        acc = fmaf(a.x, w.x, acc); acc = fmaf(a.y, w.y, acc);
        acc = fmaf(a.z, w.z, acc); acc = fmaf(a.w, w.w, acc);
    }
    #pragma unroll
    for (int s = 16; s > 0; s >>= 1) acc += __shfl_xor(acc, s, 32);
    if (lane == 0) out[(size_t)bo * C_ + j] = acc + proj_b[j];
}

// ------- Phase 6: attn_weights[b,t] = mean_h attn[b,h,t] -------
__global__ void attnw_kernel(const float* __restrict__ attn, float* __restrict__ out2) {
    const int idx = blockIdx.x * 256 + threadIdx.x;   // b*T + t
    const int b = idx / T_, t = idx % T_;
    float s = 0.f;
    #pragma unroll
    for (int h = 0; h < H_; ++h) s += attn[((size_t)b * H_ + h) * T_ + t];
    out2[idx] = s * 0.125f;
}

extern "C" void kernel_launch(void* const* d_in, const int* in_sizes, int n_in,
                              void* d_out, int out_size, void* d_ws, size_t ws_size,
                              hipStream_t stream) {
    const float* x      = (const float*)d_in[0];
    const int*   mask   = (const int*)  d_in[1];
    const float* qt     = (const float*)d_in[2];
    const float* kv_w   = (const float*)d_in[3];
    const float* kv_b   = (const float*)d_in[4];
    const float* q_w    = (const float*)d_in[5];
    const float* q_b    = (const float*)d_in[6];
    const float* proj_w = (const float*)d_in[7];
    const float* proj_b = (const float*)d_in[8];
    float* out = (float*)d_out;                 // [0 : B*C) pooled, [B*C : B*C+B*T) attn_weights

    float* ws = (float*)d_ws;
    float* w_q    = ws;                         // 1024
    float* w_qk   = ws + 1024;                  // 8192
    float* w_ck   = ws + 9216;                  // 8 (padded to 16)
    float* w_attn = ws + 9232;                  // B*H*T = 524288 (logits, softmaxed in place)
    float* w_xp   = ws + 9232 + (size_t)B_ * H_ * T_;            // B*H*C = 262144
    float* w_pool = w_xp + (size_t)B_ * H_ * C_;                 // B*C   = 32768

    q_kernel      <<<128, 256, 0, stream>>>(qt, q_w, q_b, w_q);
    qkvec_kernel  <<<32, 256, 0, stream>>>(w_q, kv_w, kv_b, w_qk, w_ck);
    logits_kernel <<<dim3(T_ / 64, B_), 256, 0, stream>>>(x, mask, w_qk, w_ck, w_attn);
    softmax_kernel<<<B_ * H_, 256, 0, stream>>>(w_attn);
    xp_wmma_kernel<<<dim3(B_, C_ / NT), 256, 0, stream>>>(w_attn, x, w_xp);
    pooled_kernel <<<(B_ * C_) / 8, 256, 0, stream>>>(w_xp, kv_w, kv_b, w_pool);
    proj_kernel   <<<(B_ * C_) / 8, 256, 0, stream>>>(w_pool, proj_w, proj_b, out);
    attnw_kernel  <<<(B_ * T_) / 256, 256, 0, stream>>>(w_attn, out + (size_t)B_ * C_);
}